// PCGSolver_21921513079015
// MI455X (gfx1250) — compile-verified
//
#include <hip/hip_runtime.h>
#include <cstddef>

// Conjugate Gradient, N=4096, 20 iterations, fp32.
// Core op: Ap = A @ p (64 MB of A per matvec). A fits in MI455X's 192 MB L2,
// so iterations 2..20 stream from L2; iteration 1 from HBM @23.3 TB/s (~2.8us).
// Matvec uses V_WMMA_F32_16X16X4_F32 with p broadcast across B's 16 columns,
// so every D column equals the GEMV result; column 0 is extracted via LDS.

#define CG_N     4096
#define CG_ITERS 20
#define CG_TOL2  1e-16f   // TOL^2, guard replicates the reference's ||r|| >= 1e-8

typedef float v2f __attribute__((ext_vector_type(2)));
typedef float v8f __attribute__((ext_vector_type(8)));

// ---------------- matvec: Ap = A @ p -----------------------------------------
// grid = 256 blocks (16 rows each), block = 256 threads = 8 waves.
// Wave w covers K in [w*512, w*512+512). 128 WMMAs per wave, 4-way acc unroll.
__global__ __launch_bounds__(256) void cg_matvec(const float* __restrict__ A,
                                                 const float* __restrict__ p,
                                                 float* __restrict__ Ap,
                                                 const float* __restrict__ scal) {
  if (scal[0] < CG_TOL2) return;  // uniform: whole wave exits, EXEC stays full
  __shared__ float red[8 * 256];
  const int tid  = threadIdx.x;
  const int wave = tid >> 5;
  const int lane = tid & 31;
  const int hi   = lane >> 4;               // 0 -> K{0,1}, 1 -> K{2,3} (ISA A/B layout)
  const int row  = (blockIdx.x << 4) + (lane & 15);
  const float* Arow = A + (size_t)row * CG_N + (hi << 1);
  const float* pv   = p + (hi << 1);
  const int k0 = wave << 9;                 // 512 K-values per wave

  v8f acc0 = {}, acc1 = {}, acc2 = {}, acc3 = {};
#pragma unroll 2
  for (int k = k0; k < k0 + 512; k += 16) {
    __builtin_prefetch(Arow + k + 256);     // global_prefetch_b8, helps HBM pass
    v2f a0 = *(const v2f*)(Arow + k);
    v2f b0 = *(const v2f*)(pv + k);
    v2f a1 = *(const v2f*)(Arow + k + 4);
    v2f b1 = *(const v2f*)(pv + k + 4);
    v2f a2 = *(const v2f*)(Arow + k + 8);
    v2f b2 = *(const v2f*)(pv + k + 8);
    v2f a3 = *(const v2f*)(Arow + k + 12);
    v2f b3 = *(const v2f*)(pv + k + 12);
    acc0 = __builtin_amdgcn_wmma_f32_16x16x4_f32(false, a0, false, b0, (short)0, acc0, false, false);
    acc1 = __builtin_amdgcn_wmma_f32_16x16x4_f32(false, a1, false, b1, (short)0, acc1, false, false);
    acc2 = __builtin_amdgcn_wmma_f32_16x16x4_f32(false, a2, false, b2, (short)0, acc2, false, false);
    acc3 = __builtin_amdgcn_wmma_f32_16x16x4_f32(false, a3, false, b3, (short)0, acc3, false, false);
  }
  v8f acc = (acc0 + acc1) + (acc2 + acc3);

  // D layout: VGPR j = row j (lanes 0-15) / row j+8 (lanes 16-31); all N cols equal.
  float* myred = red + wave * 256 + lane;
#pragma unroll
  for (int j = 0; j < 8; ++j) myred[j * 32] = acc[j];
  __syncthreads();
  if (tid < 16) {
    const int vg = tid & 7;
    const int ln = (tid >> 3) << 4;   // column 0 lives in lane 0 (rows 0-7) / lane 16 (rows 8-15)
    float s = 0.f;
#pragma unroll
    for (int w = 0; w < 8; ++w) s += red[w * 256 + vg * 32 + ln];
    Ap[(blockIdx.x << 4) + tid] = s;
  }
}

// ---------------- init: x=0, r=p=b, scal[0]=b.b ------------------------------
__global__ __launch_bounds__(1024) void cg_init(const float* __restrict__ b,
                                                float* __restrict__ x,
                                                float* __restrict__ r,
                                                float* __restrict__ p,
                                                float* __restrict__ scal) {
  __shared__ float sred[1024];
  const int tid = threadIdx.x;
  float s = 0.f;
  for (int i = tid; i < CG_N; i += 1024) {
    float v = b[i];
    x[i] = 0.f; r[i] = v; p[i] = v;
    s += v * v;
  }
  sred[tid] = s;
  __syncthreads();
  for (int off = 512; off > 0; off >>= 1) {
    if (tid < off) sred[tid] += sred[tid + off];
    __syncthreads();
  }
  if (tid == 0) { scal[0] = sred[0]; scal[1] = 0.f; scal[2] = 0.f; }
}

// ---------------- dots: scal[1] = p . Ap -------------------------------------
__global__ __launch_bounds__(1024) void cg_dot_pAp(const float* __restrict__ p,
                                                   const float* __restrict__ Ap,
                                                   float* __restrict__ scal) {
  if (scal[0] < CG_TOL2) return;
  __shared__ float sred[1024];
  const int tid = threadIdx.x;
  float s = 0.f;
  for (int i = tid; i < CG_N; i += 1024) s += p[i] * Ap[i];
  sred[tid] = s;
  __syncthreads();
  for (int off = 512; off > 0; off >>= 1) {
    if (tid < off) sred[tid] += sred[tid + off];
    __syncthreads();
  }
  if (tid == 0) scal[1] = sred[0];
}

// ---------------- update1: x += a p; r -= a Ap; scal[2] = r'.r' --------------
__global__ __launch_bounds__(1024) void cg_update_xr(float* __restrict__ x,
                                                     float* __restrict__ r,
                                                     const float* __restrict__ p,
                                                     const float* __restrict__ Ap,
                                                     float* __restrict__ scal) {
  if (scal[0] < CG_TOL2) return;
  const float a = scal[0] / scal[1];
  __shared__ float sred[1024];
  const int tid = threadIdx.x;
  float s = 0.f;
  for (int i = tid; i < CG_N; i += 1024) {
    x[i] += a * p[i];
    float rn = r[i] - a * Ap[i];
    r[i] = rn;
    s += rn * rn;
  }
  sred[tid] = s;
  __syncthreads();
  for (int off = 512; off > 0; off >>= 1) {
    if (tid < off) sred[tid] += sred[tid + off];
    __syncthreads();
  }
  if (tid == 0) scal[2] = sred[0];
}

// ---------------- update2: p = r + b p; scal[0] = scal[2] --------------------
__global__ __launch_bounds__(1024) void cg_update_p(const float* __restrict__ r,
                                                    float* __restrict__ p,
                                                    float* __restrict__ scal) {
  const float rr_old = scal[0];
  if (rr_old < CG_TOL2) return;
  const float rr_new = scal[2];
  const float bk = rr_new / rr_old;
  const int tid = threadIdx.x;
  for (int i = tid; i < CG_N; i += 1024) p[i] = r[i] + bk * p[i];
  __syncthreads();   // all threads have read scal[0] before the rotation below
  if (tid == 0) scal[0] = rr_new;
}

extern "C" void kernel_launch(void* const* d_in, const int* in_sizes, int n_in,
                              void* d_out, int out_size, void* d_ws, size_t ws_size,
                              hipStream_t stream) {
  const float* A = (const float*)d_in[0];   // [4096, 4096] fp32
  const float* b = (const float*)d_in[1];   // [4096] fp32

  float* x    = (float*)d_out;              // solution lives directly in d_out
  float* scal = (float*)d_ws;               // [0]=r.r  [1]=p.Ap  [2]=r'.r'
  float* r    = scal + 16;
  float* p    = r + CG_N;
  float* Ap   = p + CG_N;

  cg_init<<<1, 1024, 0, stream>>>(b, x, r, p, scal);
  for (int it = 0; it < CG_ITERS; ++it) {
    cg_matvec<<<CG_N / 16, 256, 0, stream>>>(A, p, Ap, scal);
    cg_dot_pAp<<<1, 1024, 0, stream>>>(p, Ap, scal);
    cg_update_xr<<<1, 1024, 0, stream>>>(x, r, p, Ap, scal);
    cg_update_p<<<1, 1024, 0, stream>>>(r, p, scal);
  }
}